// RecurrentHedgeModel_9887014716011
// MI455X (gfx1250) — compile-verified
//
#include <hip/hip_runtime.h>

// RecurrentHedgeModel on gfx1250 (MI455X).
// h0 is constant zero in the reference, so each step per batch row b is
//   pre[j] = W1[j,0]*s0 + W1[j,1]*s1 + W1[j,2]*delta_prev + 1.0*b1[j]
//   delta  = b2 + sum_j W2[j] * relu(pre[j])
// Per 16 batch rows/step: 16x4 @ 4x16 fp32 WMMA, 8 hidden tiles:
//   A_n (constant): rows = hidden units j, K = {W1[:,0],W1[:,1],W1[:,2],b1}
//   B   (per step): rows K = {s0, s1, delta_prev, 1}, cols = batch
// D_n lands with lane = batch, VGPR = hidden unit -> lane-local reduction.
// relu(x) = (x+|x|)/2 splits the output dot into
//   (a) a linear part, folded analytically into q = sum_j (w_j/2)*W1-row_j
//       (3 FMAs/step), and
//   (b) sum_j (w_j/2)*|pre_j|  ->  64 v_fma with free |abs| source modifier.

typedef float v2f __attribute__((ext_vector_type(2)));
typedef float v8f __attribute__((ext_vector_type(8)));

#define STEPS 60
#define INFEAT 131   // 2 + 1 + 128

__global__ __launch_bounds__(256) void
hedge_wmma_kernel(const float* __restrict__ S,
                  const float* __restrict__ W1,
                  const float* __restrict__ b1,
                  const float* __restrict__ W2,
                  const float* __restrict__ b2,
                  float* __restrict__ out) {
    const int lane    = threadIdx.x & 31;
    const int wave    = threadIdx.x >> 5;
    const int nl      = lane & 15;         // batch column within the wave's 16
    const bool lowhlf = (lane < 16);

    const int b0 = (blockIdx.x * 8 + wave) * 16;   // first batch row of this wave
    const int bl = b0 + nl;                        // batch row tracked by this lane

    // ---- A-matrix weight tiles (16x4 f32 each => 2 VGPRs/lane) -------------
    // A-layout: lanes0-15 rows M=j' with VGPR0=K0,VGPR1=K1;
    //           lanes16-31 rows M=j' with VGPR0=K2,VGPR1=K3.
    v2f At[8];
#pragma unroll
    for (int n = 0; n < 8; ++n) {
        const int j = 16 * n + nl;
        if (lowhlf) {
            At[n].x = W1[(size_t)j * INFEAT + 0];   // K0 = W1[:,0]
            At[n].y = W1[(size_t)j * INFEAT + 1];   // K1 = W1[:,1]
        } else {
            At[n].x = W1[(size_t)j * INFEAT + 2];   // K2 = W1[:,2]
            At[n].y = b1[j];                        // K3 = b1
        }
    }

    // ---- lane-mapped W2/2: D_n VGPR r holds j = 16n + r + 8*(lane>=16) -----
    const int hi8 = lowhlf ? 0 : 8;
    float w2h[64];
#pragma unroll
    for (int n = 0; n < 8; ++n) {
        const float4 w2a = *(const float4*)(W2 + 16 * n + hi8);
        const float4 w2b = *(const float4*)(W2 + 16 * n + hi8 + 4);
        w2h[8 * n + 0] = 0.5f * w2a.x; w2h[8 * n + 1] = 0.5f * w2a.y;
        w2h[8 * n + 2] = 0.5f * w2a.z; w2h[8 * n + 3] = 0.5f * w2a.w;
        w2h[8 * n + 4] = 0.5f * w2b.x; w2h[8 * n + 5] = 0.5f * w2b.y;
        w2h[8 * n + 6] = 0.5f * w2b.z; w2h[8 * n + 7] = 0.5f * w2b.w;
    }

    // ---- q = sum_j (W2[j]/2) * (W1[j,0], W1[j,1], W1[j,2], b1[j]) ----------
    // Cooperative: each lane covers j = lane, lane+32, lane+64, lane+96,
    // then a full-wave butterfly gives every lane the totals.
    float qx = 0.f, qy = 0.f, qz = 0.f, qw = 0.f;
#pragma unroll
    for (int it = 0; it < 4; ++it) {
        const int j = lane + 32 * it;
        const float wj = 0.5f * W2[j];
        qx = __builtin_fmaf(wj, W1[(size_t)j * INFEAT + 0], qx);
        qy = __builtin_fmaf(wj, W1[(size_t)j * INFEAT + 1], qy);
        qz = __builtin_fmaf(wj, W1[(size_t)j * INFEAT + 2], qz);
        qw = __builtin_fmaf(wj, b1[j], qw);
    }
#pragma unroll
    for (int m = 1; m < 32; m <<= 1) {
        qx += __shfl_xor(qx, m);
        qy += __shfl_xor(qy, m);
        qz += __shfl_xor(qz, m);
        qw += __shfl_xor(qw, m);
    }
    qw += b2[0];                           // fold output bias into the constant

    // ---- Recurrence over time ---------------------------------------------
    const float2* srow = (const float2*)(S + (size_t)bl * (STEPS * 2));
    float2 sc = srow[0];                   // one-step-lookahead S fetch
    float  dprev = 0.0f;

#pragma unroll 2
    for (int t = 0; t < STEPS; ++t) {
        const float2 scur = sc;            // this step's inputs (all lanes)
        // B (4x16 f32): VGPR0 = l<16 ? s0 : delta_prev ; VGPR1 = l<16 ? s1 : 1
        v2f Bv;
        Bv.x = lowhlf ? scur.x : dprev;
        Bv.y = lowhlf ? scur.y : 1.0f;
        if (t + 1 < STEPS) sc = srow[t + 1];

        // D_n = A_n @ B ; accumulate (w/2)*|pre| with free abs modifier.
        float acc[8] = {0.f, 0.f, 0.f, 0.f, 0.f, 0.f, 0.f, 0.f};
#pragma unroll
        for (int n = 0; n < 8; ++n) {
            v8f cz = {0.f, 0.f, 0.f, 0.f, 0.f, 0.f, 0.f, 0.f};
            v8f p = __builtin_amdgcn_wmma_f32_16x16x4_f32(
                        false, At[n], false, Bv, (short)0, cz, false, false);
#pragma unroll
            for (int r = 0; r < 8; ++r)
                acc[r] = __builtin_fmaf(__builtin_fabsf(p[r]),
                                        w2h[8 * n + r], acc[r]);
        }

        // Lane-local tree (64 terms) + other half's 64 terms (one swizzle).
        const float s01 = acc[0] + acc[1], s23 = acc[2] + acc[3];
        const float s45 = acc[4] + acc[5], s67 = acc[6] + acc[7];
        const float part = (s01 + s23) + (s45 + s67);
        const float asum = part + __shfl_xor(part, 16);

        // Linear half: q . (s0, s1, delta_prev, 1) (bias folded into qw).
        const float lin = __builtin_fmaf(qx, scur.x,
                          __builtin_fmaf(qy, scur.y,
                          __builtin_fmaf(qz, dprev, qw)));

        const float dnew = asum + lin;

        if (lowhlf) out[(size_t)bl * STEPS + t] = dnew;  // [B, T] layout
        dprev = dnew;                       // lane 16+b carries delta[b] into B
    }
}

extern "C" void kernel_launch(void* const* d_in, const int* in_sizes, int n_in,
                              void* d_out, int out_size, void* d_ws, size_t ws_size,
                              hipStream_t stream) {
    const float* S  = (const float*)d_in[0];
    const float* W1 = (const float*)d_in[1];
    const float* b1 = (const float*)d_in[2];
    const float* W2 = (const float*)d_in[3];
    const float* b2 = (const float*)d_in[4];
    float* out = (float*)d_out;

    const int B = in_sizes[0] / (STEPS * 2);   // 262144
    const int blocks = B / (16 * 8);           // 16 rows/wave * 8 waves/block
    hedge_wmma_kernel<<<blocks, 256, 0, stream>>>(S, W1, b1, W2, b2, out);
}